// MovingWindowHyperbolicForecaster_66649302499512
// MI455X (gfx1250) — compile-verified
//
#include <hip/hip_runtime.h>
#include <math.h>

// ---------------- constants from the reference ----------------
#define BB 512
#define LL 720
#define FF 21
#define SEG 24
#define EMBED 256
#define HID 1024
#define N_SEG 30          // L/SEG
#define N_PRED 8          // PRED_LEN/SEG
#define SF (SEG*FF)       // 504
#define ROWS_E (BB*N_SEG) // 15360 rows per component for embed GEMM
#define ROWS_S (5*BB)     // 2560 rows in the scan
#define SLOTS (N_SEG + N_PRED) // 38 window slots
#define EPS 1e-7f
#define MAXN (1.0f - 1e-5f)

#define KCH 64            // K chunk
#define ASTR 68           // LDS A row stride (x4B = 16B aligned rows; banks 4m+kb vs 4m+kb+2 disjoint)
#define BSTR 136          // LDS B row stride: 2*136 mod 64 == 16 -> half-wave groups 16 banks apart

typedef __attribute__((ext_vector_type(2))) float v2f;
typedef __attribute__((ext_vector_type(8))) float v8f;

// ---------------- helpers ----------------
__device__ __forceinline__ float atanh_f(float x) {
    // x is pre-clamped to < 1-1e-7
    return 0.5f * __logf((1.f + x) / (1.f - x));
}

// block-wide sum over 256 threads (8 wave32s)
__device__ __forceinline__ float block_reduce_sum(float v, float* sm) {
    #pragma unroll
    for (int off = 16; off > 0; off >>= 1)
        v += __shfl_down(v, off, 32);
    const int wave = threadIdx.x >> 5;
    const int lane = threadIdx.x & 31;
    if (lane == 0) sm[wave] = v;
    __syncthreads();
    if (threadIdx.x < 8) {
        float r = sm[threadIdx.x];
        #pragma unroll
        for (int off = 4; off > 0; off >>= 1)
            r += __shfl_down(r, off, 8);
        if (threadIdx.x == 0) sm[8] = r;
    }
    __syncthreads();
    float out = sm[8];
    __syncthreads();
    return out;
}

// CDNA5 async global->LDS copy (no VGPR staging, tracked by ASYNCcnt)
__device__ __forceinline__ void async_g2l_b128(const float* gp, float* lp) {
    const unsigned int lds = (unsigned int)(size_t)lp;   // low 32 bits = LDS offset
    asm volatile("global_load_async_to_lds_b128 %0, %1, off"
                 :: "v"(lds), "v"(gp) : "memory");
}
__device__ __forceinline__ void wait_async_10() {
    asm volatile("s_wait_asynccnt 0xa" ::: "memory");
}
__device__ __forceinline__ void wait_async_0() {
    asm volatile("s_wait_asynccnt 0x0" ::: "memory");
}

// ---------------- WMMA compute core over one staged 64-K chunk ----------------
__device__ __forceinline__ void wmma_chunk(const float* As, const float* Bs,
                                           int mrow, int khalf, int nloc,
                                           v8f& acc0, v8f& acc1) {
    #pragma unroll
    for (int kk = 0; kk < KCH; kk += 4) {
        const int kb = kk + 2 * khalf;
        v2f a0, a1, b;
        a0.x = As[mrow * ASTR + kb];
        a0.y = As[mrow * ASTR + kb + 1];
        a1.x = As[(mrow + 16) * ASTR + kb];
        a1.y = As[(mrow + 16) * ASTR + kb + 1];
        b.x  = Bs[kb * BSTR + nloc];
        b.y  = Bs[(kb + 1) * BSTR + nloc];
        acc0 = __builtin_amdgcn_wmma_f32_16x16x4_f32(
                   false, a0, false, b, (short)0, acc0, false, false);
        acc1 = __builtin_amdgcn_wmma_f32_16x16x4_f32(
                   false, a1, false, b, (short)0, acc1, false, false);
    }
}

// ---------------- async double-buffered fp32 WMMA GEMM ----------------
// Requires: M % 32 == 0, N % 128 == 0, K % 64 == 0 (the MLP GEMMs).
// Block = 256 threads (8 wave32s). Block tile 32(M) x 128(N).
// Each chunk: every wave issues exactly 10 async b128 copies (2 for A, 8 for B),
// so waiting ASYNCcnt <= 10 retires the previous chunk (async loads complete in order).
__global__ __launch_bounds__(256)
void gemm_wmma_f32_async(const float* __restrict__ A, const float* __restrict__ W,
                         const float* __restrict__ bias, float* __restrict__ C,
                         int M, int N, int K, int relu)
{
    __shared__ float As[2][32 * ASTR];
    __shared__ float Bs[2][KCH * BSTR];

    const int tid   = threadIdx.x;
    const int wave  = tid >> 5;
    const int lane  = tid & 31;
    const int m0    = blockIdx.y * 32;
    const int bn0   = blockIdx.x * 128;
    const int nloc  = wave * 16 + (lane & 15);
    const int mrow  = lane & 15;
    const int khalf = lane >> 4;

    const int nch = K >> 6;

    // issue chunk 0
    {
        #pragma unroll
        for (int i = tid; i < 32 * 16; i += 256) {
            const int rr = i >> 4, c4 = (i & 15) << 2;
            async_g2l_b128(&A[(size_t)(m0 + rr) * K + c4], &As[0][rr * ASTR + c4]);
        }
        #pragma unroll
        for (int i = tid; i < KCH * 32; i += 256) {
            const int kk = i >> 5, c4 = (i & 31) << 2;
            async_g2l_b128(&W[(size_t)kk * N + bn0 + c4], &Bs[0][kk * BSTR + c4]);
        }
    }

    v8f acc0 = {};
    v8f acc1 = {};

    for (int ch = 0; ch < nch; ++ch) {
        const int cur = ch & 1;
        if (ch + 1 < nch) {
            const int k0 = (ch + 1) << 6;
            #pragma unroll
            for (int i = tid; i < 32 * 16; i += 256) {
                const int rr = i >> 4, c4 = (i & 15) << 2;
                async_g2l_b128(&A[(size_t)(m0 + rr) * K + k0 + c4],
                               &As[1 - cur][rr * ASTR + c4]);
            }
            #pragma unroll
            for (int i = tid; i < KCH * 32; i += 256) {
                const int kk = i >> 5, c4 = (i & 31) << 2;
                async_g2l_b128(&W[(size_t)(k0 + kk) * N + bn0 + c4],
                               &Bs[1 - cur][kk * BSTR + c4]);
            }
            wait_async_10();          // oldest 10 (= chunk ch) have landed
        } else {
            wait_async_0();
        }
        __syncthreads();              // chunk ch visible to every wave

        wmma_chunk(As[cur], Bs[cur], mrow, khalf, nloc, acc0, acc1);

        __syncthreads();              // done reading buf before it is refilled
    }

    const int gn = bn0 + nloc;
    const float bv = bias[gn];
    #pragma unroll
    for (int v = 0; v < 8; ++v) {
        float v0 = acc0[v] + bv;
        float v1 = acc1[v] + bv;
        if (relu) { v0 = fmaxf(v0, 0.f); v1 = fmaxf(v1, 0.f); }
        C[(size_t)(m0 + v + 8 * khalf) * N + gn]      = v0;
        C[(size_t)(m0 + 16 + v + 8 * khalf) * N + gn] = v1;
    }
}

// ---------------- generic fp32 WMMA GEMM (handles ragged K / N tails) --------
__global__ __launch_bounds__(256)
void gemm_wmma_f32(const float* __restrict__ A, const float* __restrict__ W,
                   const float* __restrict__ bias, float* __restrict__ C,
                   int M, int N, int K, int relu)
{
    __shared__ float As[32 * ASTR];
    __shared__ float Bs[KCH * BSTR];

    const int tid   = threadIdx.x;
    const int wave  = tid >> 5;
    const int lane  = tid & 31;
    const int m0    = blockIdx.y * 32;
    const int bn0   = blockIdx.x * 128;
    const int nloc  = wave * 16 + (lane & 15);
    const int mrow  = lane & 15;
    const int khalf = lane >> 4;

    const bool fullN = (bn0 + 128 <= N);

    v8f acc0 = {};
    v8f acc1 = {};

    for (int k0 = 0; k0 < K; k0 += KCH) {
        const bool fullK = (k0 + KCH <= K);

        if (fullK) {
            #pragma unroll
            for (int i = tid; i < 32 * 16; i += 256) {
                const int rr = i >> 4;
                const int c4 = (i & 15) << 2;
                const float* gp = &A[(size_t)(m0 + rr) * K + k0 + c4];
                const float4 v = *(const float4*)gp;
                __builtin_prefetch(gp + KCH, 0, 1);
                As[rr * ASTR + c4 + 0] = v.x;
                As[rr * ASTR + c4 + 1] = v.y;
                As[rr * ASTR + c4 + 2] = v.z;
                As[rr * ASTR + c4 + 3] = v.w;
            }
        } else {
            for (int i = tid; i < 32 * KCH; i += 256) {
                const int rr = i >> 6;
                const int cc = i & 63;
                const int gk = k0 + cc;
                As[rr * ASTR + cc] = (gk < K) ? A[(size_t)(m0 + rr) * K + gk] : 0.f;
            }
        }

        if (fullK && fullN) {
            #pragma unroll
            for (int i = tid; i < KCH * 32; i += 256) {
                const int kk = i >> 5;
                const int c4 = (i & 31) << 2;
                const float* gp = &W[(size_t)(k0 + kk) * N + bn0 + c4];
                const float4 v = *(const float4*)gp;
                __builtin_prefetch(gp + (size_t)KCH * N, 0, 1);
                Bs[kk * BSTR + c4 + 0] = v.x;
                Bs[kk * BSTR + c4 + 1] = v.y;
                Bs[kk * BSTR + c4 + 2] = v.z;
                Bs[kk * BSTR + c4 + 3] = v.w;
            }
        } else {
            for (int i = tid; i < KCH * 128; i += 256) {
                const int kk = i >> 7;
                const int nn = i & 127;
                const int gk = k0 + kk;
                const int gn = bn0 + nn;
                Bs[kk * BSTR + nn] =
                    (gk < K && gn < N) ? W[(size_t)gk * N + gn] : 0.f;
            }
        }
        __syncthreads();

        wmma_chunk(As, Bs, mrow, khalf, nloc, acc0, acc1);

        __syncthreads();
    }

    const int gn = bn0 + nloc;
    if (gn < N) {
        const float bv = bias[gn];
        #pragma unroll
        for (int v = 0; v < 8; ++v) {
            float v0 = acc0[v] + bv;
            float v1 = acc1[v] + bv;
            if (relu) { v0 = fmaxf(v0, 0.f); v1 = fmaxf(v1, 0.f); }
            C[(size_t)(m0 + v + 8 * khalf) * N + gn]      = v0;
            C[(size_t)(m0 + 16 + v + 8 * khalf) * N + gn] = v1;
        }
    }
}

// ---------------- embed post: layernorm + expmap0 + combine ----------------
__global__ __launch_bounds__(256)
void embed_post_kernel(const float* __restrict__ tan4,
                       const float* __restrict__ gamma,
                       const float* __restrict__ beta,
                       float* __restrict__ zhist)
{
    __shared__ float sm[16];
    const int r = blockIdx.x;      // b*N_SEG + n
    const int d = threadIdx.x;
    const int b = r / N_SEG;
    const int n = r % N_SEG;

    float ysum = 0.f;
    #pragma unroll
    for (int c = 0; c < 4; ++c) {
        float t  = tan4[((size_t)c * ROWS_E + r) * EMBED + d];
        float mu = block_reduce_sum(t, sm) * (1.f / EMBED);
        float dv = t - mu;
        float var = block_reduce_sum(dv * dv, sm) * (1.f / EMBED);
        float y = dv * rsqrtf(var + 1e-5f) * gamma[c * EMBED + d] + beta[c * EMBED + d];
        ysum += y;

        float n2 = block_reduce_sum(y * y, sm);
        float nn = fmaxf(sqrtf(n2), EPS);
        float th = tanhf(nn);
        float sc = (th > MAXN) ? (MAXN / nn) : (th / nn);
        const size_t row = (size_t)(c + 1) * BB + b;
        zhist[(row * SLOTS + n) * EMBED + d] = y * sc;
    }
    float n2 = block_reduce_sum(ysum * ysum, sm);
    float nn = fmaxf(sqrtf(n2), EPS);
    float th = tanhf(nn);
    float sc = (th > MAXN) ? (MAXN / nn) : (th / nn);
    zhist[((size_t)b * SLOTS + n) * EMBED + d] = ysum * sc;
}

// ---------------- feature build: vel (29 pair logmaps) + logmap0(last/prev) ----
__global__ __launch_bounds__(256)
void feat_kernel(const float* __restrict__ zhist, float* __restrict__ feat, int t)
{
    __shared__ float sm[16];
    const int r = blockIdx.x;
    const int d = threadIdx.x;
    const float* zw = zhist + ((size_t)r * SLOTS + t) * EMBED;

    float velacc = 0.f;
    float x  = zw[d];
    float x2 = block_reduce_sum(x * x, sm);

    for (int j = 0; j < N_SEG - 1; ++j) {
        float y  = zw[(size_t)(j + 1) * EMBED + d];
        float y2 = block_reduce_sum(y * y, sm);
        float xy = block_reduce_sum(x * y, sm);
        float num = (1.f - 2.f * xy + y2) * (-x) + (1.f - x2) * y;
        float den = 1.f - 2.f * xy + x2 * y2;
        float u   = num / fmaxf(den, EPS);
        float u2  = block_reduce_sum(u * u, sm);
        float nn  = fmaxf(sqrtf(u2), EPS);
        float coef = fmaxf(1.f - x2, EPS) * atanh_f(fminf(nn, 1.f - 1e-7f)) / nn;
        velacc += coef * u;
        x = y; x2 = y2;
    }
    float vel = velacc * (1.f / (N_SEG - 1));

    float nl  = fmaxf(sqrtf(x2), EPS);
    float lml = atanh_f(fminf(nl, 1.f - 1e-7f)) / nl * x;

    float zp  = zw[(size_t)(N_SEG - 2) * EMBED + d];
    float zp2 = block_reduce_sum(zp * zp, sm);
    float np  = fmaxf(sqrtf(zp2), EPS);
    float lmp = atanh_f(fminf(np, 1.f - 1e-7f)) / np * zp;

    float* fr = feat + (size_t)r * (3 * EMBED);
    fr[d]             = lml;
    fr[EMBED + d]     = lmp;
    fr[2 * EMBED + d] = vel;
}

// ---------------- z_next = expmap(z_last, mlp_out); write next slot ----------
__global__ __launch_bounds__(256)
void expmap_step_kernel(float* __restrict__ zhist, const float* __restrict__ mlp, int t)
{
    __shared__ float sm[16];
    const int r = blockIdx.x;
    const int d = threadIdx.x;

    float x = zhist[((size_t)r * SLOTS + t + N_SEG - 1) * EMBED + d];
    float v = mlp[(size_t)r * EMBED + d];
    float x2 = block_reduce_sum(x * x, sm);
    float v2 = block_reduce_sum(v * v, sm);

    float nn  = fmaxf(sqrtf(v2), EPS);
    float lam = 2.f / fmaxf(1.f - x2, EPS);
    float s   = tanhf(lam * nn * 0.5f) / nn;   // SC == 1
    float y   = s * v;
    float y2  = s * s * v2;
    float xy  = s * block_reduce_sum(x * v, sm);

    float num = (1.f + 2.f * xy + y2) * x + (1.f - x2) * y;
    float den = 1.f + 2.f * xy + x2 * y2;
    float res = num / fmaxf(den, EPS);

    float r2 = block_reduce_sum(res * res, sm);
    float rn = sqrtf(r2);
    if (rn > MAXN) res = res / fmaxf(rn, EPS) * MAXN;

    zhist[((size_t)r * SLOTS + t + N_SEG) * EMBED + d] = res;
}

// ---------------- logmap0 over the 8 predicted slots -------------------------
__global__ __launch_bounds__(256)
void logmap0_kernel(const float* __restrict__ zhist, float* __restrict__ lg)
{
    __shared__ float sm[16];
    const int idx = blockIdx.x;          // r*N_PRED + t
    const int r = idx >> 3;
    const int t = idx & 7;
    const int d = threadIdx.x;

    float z  = zhist[((size_t)r * SLOTS + N_SEG + t) * EMBED + d];
    float n2 = block_reduce_sum(z * z, sm);
    float nn = fmaxf(sqrtf(n2), EPS);
    float a  = atanh_f(fminf(nn, 1.f - 1e-7f)) / nn;
    lg[(size_t)idx * EMBED + d] = a * z;
}

// ---------------- host driver ----------------
extern "C" void kernel_launch(void* const* d_in, const int* in_sizes, int n_in,
                              void* d_out, int out_size, void* d_ws, size_t ws_size,
                              hipStream_t stream) {
    const float* comp[4] = { (const float*)d_in[0], (const float*)d_in[1],
                             (const float*)d_in[2], (const float*)d_in[3] };
    const float* W_embed = (const float*)d_in[4];
    const float* b_embed = (const float*)d_in[5];
    const float* ln_g    = (const float*)d_in[6];
    const float* ln_b    = (const float*)d_in[7];
    const float* W1 = (const float*)d_in[8];
    const float* b1 = (const float*)d_in[9];
    const float* W2 = (const float*)d_in[10];
    const float* b2 = (const float*)d_in[11];
    const float* W3 = (const float*)d_in[12];
    const float* b3 = (const float*)d_in[13];
    const float* W_out = (const float*)d_in[14];
    const float* b_out = (const float*)d_in[15];

    float* ws = (float*)d_ws;
    float* zhist = ws;                                        // 2560*38*256
    float* tanb  = ws + (size_t)ROWS_S * SLOTS * EMBED;       // 4*15360*256 scratch region
    float* feat  = tanb;                                      // reuse after embed_post
    float* h1    = feat + (size_t)ROWS_S * 3 * EMBED;
    float* h2    = h1   + (size_t)ROWS_S * HID;
    float* mlp   = h2   + (size_t)ROWS_S * HID;
    float* lg    = tanb;                                      // reuse after scan

    dim3 blk(256);

    // Embed GEMMs: (15360 x 504) @ (504 x 256) per component (ragged K -> sync kernel)
    for (int c = 0; c < 4; ++c) {
        gemm_wmma_f32<<<dim3(2, ROWS_E / 32), blk, 0, stream>>>(
            comp[c], W_embed + (size_t)c * SF * EMBED, b_embed + c * EMBED,
            tanb + (size_t)c * ROWS_E * EMBED, ROWS_E, EMBED, SF, 0);
    }
    embed_post_kernel<<<ROWS_E, blk, 0, stream>>>(tanb, ln_g, ln_b, zhist);

    // Scan: 8 autoregressive steps (regular shapes -> async double-buffered kernel)
    for (int t = 0; t < N_PRED; ++t) {
        feat_kernel<<<ROWS_S, blk, 0, stream>>>(zhist, feat, t);
        gemm_wmma_f32_async<<<dim3(8, ROWS_S / 32), blk, 0, stream>>>(
            feat, W1, b1, h1, ROWS_S, HID, 3 * EMBED, 1);
        gemm_wmma_f32_async<<<dim3(8, ROWS_S / 32), blk, 0, stream>>>(
            h1, W2, b2, h2, ROWS_S, HID, HID, 1);
        gemm_wmma_f32_async<<<dim3(2, ROWS_S / 32), blk, 0, stream>>>(
            h2, W3, b3, mlp, ROWS_S, EMBED, HID, 0);
        expmap_step_kernel<<<ROWS_S, blk, 0, stream>>>(zhist, mlp, t);
    }

    // Output: logmap0 then (20480 x 256) @ (256 x 504) (ragged N -> sync kernel)
    logmap0_kernel<<<ROWS_S * N_PRED, blk, 0, stream>>>(zhist, lg);
    gemm_wmma_f32<<<dim3(4, (ROWS_S * N_PRED) / 32), blk, 0, stream>>>(
        lg, W_out, b_out, (float*)d_out, ROWS_S * N_PRED, SF, EMBED, 0);
}